// AutoEncoderGRU_20925080666567
// MI455X (gfx1250) — compile-verified
//
#include <hip/hip_runtime.h>

// AutoEncoderGRU for MI455X (gfx1250), compile-only tuned.
//
// Shapes: B=4096, T=4096, I=1, H=3. Nonlinear recurrence => 8192-step
// dependent chain per sequence; bound is chain latency, not FLOPs/BW
// (total traffic ~270MB ~= 12us at 23.3 TB/s). Mapping: lane = sequence,
// wave32, weights uniform (scalar loads), activations via v_exp_f32/v_rcp_f32.
// WMMA is wrong for the H=3 recurrence (16x16x4 min shape + per-step
// transpose would lengthen the critical chain), but right for the final
// 4096-way f32 loss reduction (row-sum trick on v_wmma_f32_16x16x4_f32).
// x is streamed per-lane with float4 double-buffering + global_prefetch_b8.

namespace {

constexpr int kB = 4096;
constexpr int kT = 4096;

typedef __attribute__((ext_vector_type(2))) float v2f;
typedef __attribute__((ext_vector_type(8))) float v8f;

__device__ __forceinline__ float sigmoid_f(float x) {
  // 1/(1+2^(-x*log2e)) : v_exp_f32 + v_rcp_f32, shortest chain.
  return __builtin_amdgcn_rcpf(1.0f + __builtin_amdgcn_exp2f(-1.4426950408889634f * x));
}
__device__ __forceinline__ float tanh_f(float x) {
  // tanh(x) = 1 - 2/(2^(2x*log2e)+1)
  return 1.0f - 2.0f * __builtin_amdgcn_rcpf(1.0f + __builtin_amdgcn_exp2f(2.8853900817779268f * x));
}

// PyTorch GRUCell step, H=3, I=1, fully unrolled per-lane scalar math.
__device__ __forceinline__ void gru_step(float x, float h[3],
                                         const float wih[9], const float whh[27],
                                         const float bih[9], const float bhh[9]) {
  float gi[9], gh[9];
#pragma unroll
  for (int g = 0; g < 9; ++g) {
    gi[g] = fmaf(x, wih[g], bih[g]);
    gh[g] = fmaf(h[0], whh[3 * g + 0],
            fmaf(h[1], whh[3 * g + 1],
            fmaf(h[2], whh[3 * g + 2], bhh[g])));
  }
#pragma unroll
  for (int c = 0; c < 3; ++c) {
    float r = sigmoid_f(gi[c] + gh[c]);
    float z = sigmoid_f(gi[3 + c] + gh[3 + c]);
    float n = tanh_f(fmaf(r, gh[6 + c], gi[6 + c]));
    h[c] = fmaf(z, h[c] - n, n);  // (1-z)*n + z*h
  }
}

__global__ __launch_bounds__(256) void gru_main(
    const float* __restrict__ x, const int* __restrict__ lens,
    const float* __restrict__ eWih, const float* __restrict__ eWhh,
    const float* __restrict__ ebih, const float* __restrict__ ebhh,
    const float* __restrict__ dWih, const float* __restrict__ dWhh,
    const float* __restrict__ dbih, const float* __restrict__ dbhh,
    const float* __restrict__ linW, const float* __restrict__ linb,
    float* __restrict__ xpad_out, float* __restrict__ out_out,
    float* __restrict__ sse_out) {
  const int b = blockIdx.x * blockDim.x + threadIdx.x;
  if (b >= kB) return;

  // ---- uniform weights -> scalar loads / SGPRs ----
  float ewih[9], ebihr[9], ebhhr[9], ewhh[27];
  float dwih[9], dbihr[9], dbhhr[9], dwhh[27];
#pragma unroll
  for (int g = 0; g < 9; ++g) {
    ewih[g] = eWih[g];  ebihr[g] = ebih[g];  ebhhr[g] = ebhh[g];
    dwih[g] = dWih[g];  dbihr[g] = dbih[g];  dbhhr[g] = dbhh[g];
  }
#pragma unroll
  for (int i = 0; i < 27; ++i) { ewhh[i] = eWhh[i]; dwhh[i] = dWhh[i]; }
  const float lw0 = linW[0], lw1 = linW[1], lw2 = linW[2], lb = linb[0];

  const int len = lens[b];
  const size_t row = (size_t)b * kT;
  const float4* __restrict__ xrow = (const float4*)(x + row);  // 16B aligned
  float* __restrict__ xprow = xpad_out + row;  // d_out+1: only 4B aligned -> scalar stores
  float* __restrict__ orow  = out_out + row;

  // ================= encoder =================
  float h[3] = {0.f, 0.f, 0.f};
  float4 cur = xrow[0];
  __builtin_prefetch((const void*)(xrow + 8), 0, 1);   // -> global_prefetch_b8
  __builtin_prefetch((const void*)(xrow + 16), 0, 1);
  for (int tb = 0; tb < kT / 4; ++tb) {
    float4 nxt = (tb + 1 < kT / 4) ? xrow[tb + 1] : make_float4(0.f, 0.f, 0.f, 0.f);
    if (((tb & 7) == 0) && (tb + 24 < kT / 4))
      __builtin_prefetch((const void*)(xrow + tb + 24), 0, 1);
    const float xs[4] = {cur.x, cur.y, cur.z, cur.w};
    const int t0 = tb * 4;
#pragma unroll
    for (int u = 0; u < 4; ++u) {
      const bool valid = (t0 + u) < len;
      const float xv = valid ? xs[u] : 0.0f;     // x_pad semantics
      xprow[t0 + u] = xv;
      float hn[3] = {h[0], h[1], h[2]};
      gru_step(xv, hn, ewih, ewhh, ebihr, ebhhr);
#pragma unroll
      for (int c = 0; c < 3; ++c) h[c] = valid ? hn[c] : h[c];  // packed-seq keep
    }
    cur = nxt;
  }

  // features = sigmoid(h_final)
  float hd[3];
#pragma unroll
  for (int c = 0; c < 3; ++c) hd[c] = sigmoid_f(h[c]);

  // ================= decoder (autoregressive) =================
  float inp = 0.0f, sse = 0.0f;
  cur = xrow[0];  // re-stream x for the loss term
  __builtin_prefetch((const void*)(xrow + 8), 0, 1);
  __builtin_prefetch((const void*)(xrow + 16), 0, 1);
  for (int tb = 0; tb < kT / 4; ++tb) {
    float4 nxt = (tb + 1 < kT / 4) ? xrow[tb + 1] : make_float4(0.f, 0.f, 0.f, 0.f);
    if (((tb & 7) == 0) && (tb + 24 < kT / 4))
      __builtin_prefetch((const void*)(xrow + tb + 24), 0, 1);
    const float xs[4] = {cur.x, cur.y, cur.z, cur.w};
    const int t0 = tb * 4;
#pragma unroll
    for (int u = 0; u < 4; ++u) {
      gru_step(inp, hd, dwih, dwhh, dbihr, dbhhr);
      const float out = fmaf(hd[0], lw0, fmaf(hd[1], lw1, fmaf(hd[2], lw2, lb)));
      const bool valid = (t0 + u) < len;
      const float outm = valid ? out : 0.0f;     // zero past seq_len
      const float xv = valid ? xs[u] : 0.0f;     // x_pad
      const float d = xv - outm;
      sse = fmaf(d, d, sse);
      orow[t0 + u] = outm;
      inp = out;  // feedback uses UN-masked out (matches reference scan carry)
    }
    cur = nxt;
  }
  sse_out[b] = sse;
}

// 4096 -> 1 reduction on the matrix pipe. Single full wave32 (EXEC all 1s).
// A tile = 16x4 chunk of partials, B = ones => D accumulates row sums in f32.
// All D columns are identical; total = sum_m D[m][0] = (lanes 0 & 16, VGPRs 0..7).
__global__ __launch_bounds__(32) void loss_reduce(const float* __restrict__ sse,
                                                  float* __restrict__ out0) {
  const int l = threadIdx.x;
  const int m = l & 15;            // A row for this lane
  const int koff = (l >> 4) * 2;   // lanes 0-15: K=0,1 ; lanes 16-31: K=2,3
  v8f c = {0.f, 0.f, 0.f, 0.f, 0.f, 0.f, 0.f, 0.f};
  const v2f ones = {1.0f, 1.0f};   // B (4x16) all-ones, any layout
  for (int i = 0; i < kB; i += 64) {
    const float* p = sse + i + m * 4 + koff;
    v2f a;
    a[0] = p[0];
    a[1] = p[1];
    c = __builtin_amdgcn_wmma_f32_16x16x4_f32(false, a, false, ones,
                                              (short)0, c, false, false);
  }
  float s = c[0] + c[1] + c[2] + c[3] + c[4] + c[5] + c[6] + c[7];
  const float tot = s + __shfl_xor(s, 16, 32);   // rows 0..7 + rows 8..15 (N=0 col)
  if (l == 0) out0[0] = tot * (1.0f / ((float)kB * (float)kT));  // mean over B*T*I
}

}  // namespace

extern "C" void kernel_launch(void* const* d_in, const int* in_sizes, int n_in,
                              void* d_out, int out_size, void* d_ws, size_t ws_size,
                              hipStream_t stream) {
  const float* x    = (const float*)d_in[0];
  const int*   lens = (const int*)d_in[1];
  const float* eWih = (const float*)d_in[2];
  const float* eWhh = (const float*)d_in[3];
  const float* ebih = (const float*)d_in[4];
  const float* ebhh = (const float*)d_in[5];
  const float* dWih = (const float*)d_in[6];
  const float* dWhh = (const float*)d_in[7];
  const float* dbih = (const float*)d_in[8];
  const float* dbhh = (const float*)d_in[9];
  const float* linW = (const float*)d_in[10];
  const float* linb = (const float*)d_in[11];

  float* out  = (float*)d_out;                       // [loss | x_pad | output]
  float* xpad = out + 1;
  float* outp = out + 1 + (size_t)kB * kT;
  float* sse  = (float*)d_ws;                        // 4096 floats (16 KB)

  // 4096 sequences, lane=sequence: 16 blocks x 256 = 128 waves,
  // 8 waves/WGP (2 per SIMD32) so transcendental latency interleaves.
  gru_main<<<dim3(16), dim3(256), 0, stream>>>(
      x, lens, eWih, eWhh, ebih, ebhh, dWih, dWhh, dbih, dbhh, linW, linb,
      xpad, outp, sse);
  loss_reduce<<<dim3(1), dim3(32), 0, stream>>>(sse, out);
}